// FocalCTCLoss_44315472560178
// MI455X (gfx1250) — compile-verified
//
#include <hip/hip_runtime.h>
#include <hip/hip_bf16.h>

typedef __attribute__((ext_vector_type(2))) float v2f;
typedef __attribute__((ext_vector_type(8))) float v8f;

#define BB 32
#define TT 512
#define CC 4000
#define SS 40
#define LL 81
#define NEGV (-1e30f)

// Exact wave32-wide sum using V_WMMA_F32_16X16X4_F32.
// A (16x4 f32): lane m (m<16) holds A[m,0] in v0, A[m,1] in v1;
// lane m+16 holds A[m,2] in v0, A[m,3] in v1.  Set v0 = value, v1 = 0.
// B = all ones -> D[m,n] = val(lane m) + val(lane m+16) for every n.
// D (16x16 f32): lane n, vgpr r = D[r,n] (r=0..7); lane n+16 = D[r+8,n].
// Sum the 8 accumulator vgprs, then one shfl_xor(16) add -> full sum in all lanes.
__device__ __forceinline__ float wave_sum_wmma(float v) {
  v2f a; a.x = v;    a.y = 0.0f;
  v2f bm; bm.x = 1.0f; bm.y = 1.0f;
  v8f c = {};
  c = __builtin_amdgcn_wmma_f32_16x16x4_f32(false, a, false, bm,
                                            (short)0, c, false, false);
  float s = ((c[0] + c[1]) + (c[2] + c[3])) + ((c[4] + c[5]) + (c[6] + c[7]));
  s += __shfl_xor(s, 16, 32);
  return s;
}

__device__ __forceinline__ float lae(float x, float y) {
  float m = fmaxf(x, y);
  float d = fminf(x, y) - m;          // <= 0; finite for our -1e30 sentinels
  return m + log1pf(__expf(d));
}

// Kernel 1: one wave per (b,t) row. Streaming online logsumexp over C=4000,
// WMMA cross-lane reduction, then cache-hot gather of the 81 extended-label
// log-probs into the lp_ext scratch buffer.
__global__ __launch_bounds__(256) void k_lse_lp(const float* __restrict__ tok,
                                                const int* __restrict__ tgt,
                                                float* __restrict__ lp) {
  const int wave = threadIdx.x >> 5;
  const int lane = threadIdx.x & 31;
  const int row  = blockIdx.x * 8 + wave;      // row = b*T + t
  const int b    = row >> 9;                   // T = 512

  const float*  base = tok + (size_t)row * CC;
  const float4* p    = (const float4*)base;    // 4000 floats = 1000 float4 (16B aligned)

  float m = -3.0e38f, s = 0.0f;
  for (int i = lane; i < CC / 4; i += 32) {
    __builtin_prefetch(p + i + 64, 0, 0);      // global_prefetch_b8
    float4 x = p[i];
    float xm = fmaxf(fmaxf(x.x, x.y), fmaxf(x.z, x.w));
    float nm = fmaxf(m, xm);
    s = s * __expf(m - nm)
      + __expf(x.x - nm) + __expf(x.y - nm)
      + __expf(x.z - nm) + __expf(x.w - nm);
    m = nm;
  }

  float M = m;
#pragma unroll
  for (int off = 16; off > 0; off >>= 1) M = fmaxf(M, __shfl_xor(M, off, 32));
  float sv  = s * __expf(m - M);
  float tot = wave_sum_wmma(sv);               // exact wave-wide sum (v_wmma)
  float lse = M + __logf(tot);                 // tot >= 1

  float* out = lp + (size_t)row * LL;
  const int* tb = tgt + b * SS;
  for (int l = lane; l < LL; l += 32) {
    int cls = (l & 1) ? tb[l >> 1] : 0;        // blank = class 0 at even slots
    out[l] = base[cls] - lse;                  // row still hot in WGP$/L2
  }
}

// Kernel 2: one wave32 per sample. 3 states per lane (81 states), LDS-shifted
// alpha recursion over T=512 steps (single-wave workgroup: barriers are HW NOPs).
__global__ __launch_bounds__(32) void k_ctc(const float* __restrict__ lp,
                                            const int* __restrict__ tgt,
                                            const int* __restrict__ tlen,
                                            float* __restrict__ losses) {
  const int b    = blockIdx.x;
  const int lane = threadIdx.x;
  __shared__ float sa[2 + 96];
  if (lane < 2) sa[lane] = NEGV;               // alpha[-1], alpha[-2] pads

  const int*   tb  = tgt + b * SS;
  const float* lpb = lp + (size_t)b * TT * LL;

  float a[3];
  bool  valid[3], skip[3];
  int   st[3];
#pragma unroll
  for (int j = 0; j < 3; ++j) {
    int s = lane + 32 * j;
    st[j]    = s;
    valid[j] = (s < LL);
    skip[j]  = false;
    if (valid[j] && (s & 1) && s >= 3) {
      int u = (s - 1) >> 1;
      skip[j] = (tb[u] != tb[u - 1]);          // targets are never blank
    }
    a[j] = (valid[j] && s < 2) ? lpb[s] : NEGV;  // alpha0
  }
  __syncthreads();

  for (int t = 1; t < TT; ++t) {
#pragma unroll
    for (int j = 0; j < 3; ++j) if (valid[j]) sa[2 + st[j]] = a[j];
    __syncthreads();
    float p1[3], p2[3];
#pragma unroll
    for (int j = 0; j < 3; ++j)
      if (valid[j]) { p1[j] = sa[1 + st[j]]; p2[j] = sa[st[j]]; }
    __syncthreads();
    const float* lpt = lpb + (size_t)t * LL;
#pragma unroll
    for (int j = 0; j < 3; ++j) {
      if (valid[j]) {
        float na = lae(a[j], p1[j]);
        if (skip[j]) na = lae(na, p2[j]);
        a[j] = na + lpt[st[j]];
      }
    }
  }

#pragma unroll
  for (int j = 0; j < 3; ++j) if (valid[j]) sa[2 + st[j]] = a[j];
  __syncthreads();

  if (lane == 0) {
    int end    = 2 * tlen[b];                  // <= 80
    float loss = -lae(sa[2 + end], sa[1 + end]);
    if (!(loss < 1e29f)) loss = 0.0f;          // also maps NaN -> 0 like jnp.where
    float pt = __expf(-loss);
    float om = 1.0f - pt;
    losses[b] = 0.25f * om * om * loss;        // ALPHA*(1-pt)^GAMMA*loss
  }
}

// Kernel 3: deterministic mean over the 32 per-sample losses.
__global__ __launch_bounds__(32) void k_final(const float* __restrict__ losses,
                                              float* __restrict__ out) {
  float v = losses[threadIdx.x];
#pragma unroll
  for (int off = 16; off > 0; off >>= 1) v += __shfl_xor(v, off, 32);
  if (threadIdx.x == 0) out[0] = v * (1.0f / 32.0f);
}

extern "C" void kernel_launch(void* const* d_in, const int* in_sizes, int n_in,
                              void* d_out, int out_size, void* d_ws, size_t ws_size,
                              hipStream_t stream) {
  const float* tok  = (const float*)d_in[0];   // (32,512,4000) f32
  const int*   tgt  = (const int*)d_in[1];     // (32,40) i32
  const int*   tlen = (const int*)d_in[2];     // (32,) i32

  float* lp     = (float*)d_ws;                        // B*T*L floats (~5.06 MB)
  float* losses = lp + (size_t)BB * TT * LL;           // +32 floats

  k_lse_lp<<<(BB * TT) / 8, 256, 0, stream>>>(tok, tgt, lp);
  k_ctc  <<<BB, 32, 0, stream>>>(lp, tgt, tlen, losses);
  k_final<<<1, 32, 0, stream>>>(losses, (float*)d_out);
}